// ROIPool_17128329576557
// MI455X (gfx1250) — compile-verified
//
#include <hip/hip_runtime.h>

// ROI adaptive max-pool (7x7) for images [8,256,56,56] f32, 256 ROIs.
// Memory-bound gather; images fit in L2 (25.7MB vs 192MB). CDNA5 path:
// one Tensor-Data-Mover DMA per block stages a 3D tile (4 channels x Hr x Wr)
// of the ROI crop into LDS (TENSORcnt), then separable max-pool from LDS.

#define ROI_R 256
#define ROI_C 256
#define IMG_N 8
#define IMG_H 56
#define IMG_W 56
#define POUT  7
#define CPB   4   // channels (waves) per block

typedef __attribute__((address_space(1))) int as1_int;
typedef __attribute__((address_space(3))) int as3_int;

typedef unsigned int u32x4 __attribute__((ext_vector_type(4)));
typedef int          i32x4 __attribute__((ext_vector_type(4)));
typedef int          i32x8 __attribute__((ext_vector_type(8)));

#if __has_builtin(__builtin_amdgcn_tensor_load_to_lds)
#define HAVE_TDM 1
#else
#define HAVE_TDM 0
#endif

__device__ __forceinline__ unsigned lds_offset(void* p) {
    // AS3 pointers are 32-bit LDS byte offsets.
    return (unsigned)(unsigned long long)(as3_int*)p;
}

__device__ __forceinline__ void async_copy_b32(const float* g, float* l) {
#if __has_builtin(__builtin_amdgcn_global_load_async_to_lds_b32)
    __builtin_amdgcn_global_load_async_to_lds_b32(
        (as1_int*)(float*)(g), (as3_int*)(l), 0, 0);
#else
    unsigned lds_off = (unsigned)(unsigned long long)l;
    asm volatile("global_load_async_to_lds_b32 %0, %1, off"
                 :: "v"(lds_off), "v"(g) : "memory");
#endif
}

__device__ __forceinline__ void wait_async0() {
#if __has_builtin(__builtin_amdgcn_s_wait_asynccnt)
    __builtin_amdgcn_s_wait_asynccnt(0);
#else
    asm volatile("s_wait_asynccnt 0" ::: "memory");
#endif
    asm volatile("" ::: "memory");
}

__device__ __forceinline__ void wait_tensor0() {
#if __has_builtin(__builtin_amdgcn_s_wait_tensorcnt)
    __builtin_amdgcn_s_wait_tensorcnt(0);
#else
    asm volatile("s_wait_tensorcnt 0" ::: "memory");
#endif
    asm volatile("" ::: "memory");
}

__global__ __launch_bounds__(CPB * 32)
void roi_adaptive_maxpool_kernel(const float* __restrict__ images,
                                 const float* __restrict__ rois,
                                 const int*   __restrict__ roi_idx,
                                 float*       __restrict__ out) {
    __shared__ float buf[CPB * IMG_H * IMG_W];   // staged crops, compact [ch][Hr][Wr]
    __shared__ float rowp[CPB][POUT * IMG_W];    // row-pooled partials

    const int lane = threadIdx.x & 31;
    const int wid  = threadIdx.x >> 5;           // wave id = local channel
    const int r    = blockIdx.x >> 6;            // ROI index (C/CPB = 64 groups)
    const int chb  = (blockIdx.x & 63) << 2;     // channel base for this block
    const int ch   = chb | wid;

    // ROI geometry (identical to reference: floor/ceil of rel coords * dim)
    const float f0 = rois[r * 4 + 0];
    const float f1 = rois[r * 4 + 1];
    const float f2 = rois[r * 4 + 2];
    const float f3 = rois[r * 4 + 3];
    int x1 = (int)floorf(f0 * (float)IMG_W);
    int y1 = (int)floorf(f1 * (float)IMG_H);
    int x2 = (int)ceilf (f2 * (float)IMG_W);
    int y2 = (int)ceilf (f3 * (float)IMG_H);
    if (x1 < 0) x1 = 0;
    if (y1 < 0) y1 = 0;
    if (x2 > IMG_W) x2 = IMG_W;
    if (y2 > IMG_H) y2 = IMG_H;
    const int Wr = x2 - x1;
    const int Hr = y2 - y1;

    float* outp = out + ((size_t)r * ROI_C + ch) * (POUT * POUT);

    if (Wr <= 0 || Hr <= 0) {   // block-uniform (depends only on r)
        for (int o = lane; o < POUT * POUT; o += 32) outp[o] = -__FLT_MAX__;
        return;
    }

    const int n = roi_idx[r];
    float* mybuf = buf + wid * (Hr * Wr);

#if HAVE_TDM
    // ---- Tensor Data Mover: one DMA stages all CPB channel crops ----
    if (wid == 0) {
        const float* src_base =
            images + ((((size_t)n * ROI_C + chb) * IMG_H + y1) * IMG_W + x1);
        const unsigned long long ga = (unsigned long long)(const void*)src_base;
        const unsigned lds = lds_offset((void*)buf);

        // D# group 0: count=1 (user), lds_addr, global_addr[56:0], type=2
        u32x4 g0;
        g0.x = 1u;
        g0.y = lds;
        g0.z = (unsigned)(ga & 0xFFFFFFFFu);
        g0.w = (unsigned)((ga >> 32) & 0x1FFFFFFu) | (2u << 30);

        // D# group 1: data_size=4B; tensor_dim0=Wr, tensor_dim1=Hr;
        // tile = Wr x Hr x CPB; stride0=W (row), stride1=H*W (channel)
        i32x8 g1;
        g1[0] = 0x00020000;                       // data_size=2 (4 bytes)
        g1[1] = (Wr & 0xFFFF) << 16;              // tensor_dim0[15:0] @ bit48
        g1[2] = (Hr & 0xFFFF) << 16;              // tensor_dim0[31:16]=0 | tensor_dim1[15:0]
        g1[3] = (Wr & 0xFFFF) << 16;              // tensor_dim1[31:16]=0 | tile_dim0=Wr
        g1[4] = (Hr & 0xFFFF) | (CPB << 16);      // tile_dim1=Hr | tile_dim2=CPB
        g1[5] = IMG_W;                            // tensor_dim0_stride[31:0] = 56
        g1[6] = (IMG_H * IMG_W) << 16;            // stride0[47:32]=0 | stride1[15:0]
        g1[7] = (IMG_H * IMG_W) >> 16;            // stride1[47:16]

        // D# group 2: tensor_dim2=CPB; no dim3. group 3: zeros.
        i32x4 g2 = {CPB, 0, 0, 0};
        i32x4 g3 = {0, 0, 0, 0};

#if __has_include(<hip/amd_detail/amd_gfx1250_TDM.h>)
        i32x8 z8 = {0, 0, 0, 0, 0, 0, 0, 0};
        __builtin_amdgcn_tensor_load_to_lds(g0, g1, g2, g3, z8, 0);   // clang-23 form
#else
        __builtin_amdgcn_tensor_load_to_lds(g0, g1, g2, g3, 0);       // ROCm 7.2 form
#endif
        wait_tensor0();
    }
    __syncthreads();
#else
    // ---- Fallback: per-lane async global->LDS b32 staging ----
    {
        const float* src =
            images + ((((size_t)n * ROI_C + ch) * IMG_H + y1) * IMG_W + x1);
        for (int y = 0; y < Hr; ++y) {
            const float* srow = src + y * IMG_W;
            float* drow = mybuf + y * Wr;
            for (int x = lane; x < Wr; x += 32)
                async_copy_b32(srow + x, drow + x);
        }
        wait_async0();
    }
    __syncthreads();
#endif

    // Row-bin max: rowp[i][x] = max over rows [floor(i*Hr/7), ceil((i+1)*Hr/7))
    float* myrow = rowp[wid];
    for (int i = 0; i < POUT; ++i) {
        const int rs = (i * Hr) / POUT;
        const int re = ((i + 1) * Hr + POUT - 1) / POUT;
        for (int x = lane; x < Wr; x += 32) {
            float m = -__FLT_MAX__;
            const float* col = mybuf + x;
            for (int y = rs; y < re; ++y) m = fmaxf(m, col[y * Wr]);
            myrow[i * IMG_W + x] = m;
        }
    }
    // Same-wave LDS ops are in-order (DScnt); lanes of this wave wrote myrow.

    // Column-bin max over the 7 row-pooled vectors -> 49 outputs.
    for (int o = lane; o < POUT * POUT; o += 32) {
        const int i  = o / POUT;
        const int j  = o - i * POUT;
        const int cs = (j * Wr) / POUT;
        const int ce = ((j + 1) * Wr + POUT - 1) / POUT;
        float m = -__FLT_MAX__;
        const float* rp = myrow + i * IMG_W;
        for (int x = cs; x < ce; ++x) m = fmaxf(m, rp[x]);
        outp[o] = m;
    }
}

extern "C" void kernel_launch(void* const* d_in, const int* in_sizes, int n_in,
                              void* d_out, int out_size, void* d_ws, size_t ws_size,
                              hipStream_t stream) {
    const float* images  = (const float*)d_in[0];
    const float* rois    = (const float*)d_in[1];
    const int*   roi_idx = (const int*)d_in[2];
    float* out = (float*)d_out;

    dim3 grid(ROI_R * (ROI_C / CPB));   // 16384 blocks
    dim3 block(CPB * 32);               // 4 wave32s
    hipLaunchKernelGGL(roi_adaptive_maxpool_kernel, grid, block, 0, stream,
                       images, rois, roi_idx, out);
}